// HungarianMatcher_57569741635899
// MI455X (gfx1250) — compile-verified
//
#include <hip/hip_runtime.h>
#include <hip/hip_bf16.h>
#include <math.h>

typedef __attribute__((ext_vector_type(2))) float v2f;
typedef __attribute__((ext_vector_type(8))) float v8f;

#define NUM_CLASSES 4
#define NOBJ 16

__global__ __launch_bounds__(256) void hungarian_cost_kernel(
    const float* __restrict__ class_probs,  // (B,16,5)
    const float* __restrict__ mask_pred,    // (B,16,L)
    const float* __restrict__ mask_tgt,     // (B,16,L)
    const float* __restrict__ reg_pred,     // (B,16,4)
    const float* __restrict__ reg_tgt,      // (B,16,4)
    const int*   __restrict__ object_class, // (B,16)
    float* __restrict__ out_cost,           // (B,16,16)
    float* __restrict__ out_len,            // (B,)
    int L)
{
  __shared__ float sMat[5][NOBJ * NOBJ];
  __shared__ float sVec[4][NOBJ];   // 0:prob_sum[n] 1:neg_sum[n] 2:fneg_sum[n] 3:tgt_sum[m]
  __shared__ int   sOC[NOBJ];

  const int b    = blockIdx.x;
  const int tid  = threadIdx.x;
  const int lane = tid & 31;
  const int wave = tid >> 5;

  // zero LDS accumulators, stage object classes
  for (int i = tid; i < 5 * NOBJ * NOBJ; i += 256) (&sMat[0][0])[i] = 0.0f;
  if (tid < 4 * NOBJ) (&sVec[0][0])[tid] = 0.0f;
  if (tid < NOBJ)     sOC[tid] = object_class[(size_t)b * NOBJ + tid];
  __syncthreads();

  // Fragment addressing: identical pattern serves both A (pred-derived) and B (tgt).
  const int row   = lane & 15;          // M for A, N for B
  const int khalf = (lane >> 4) * 2;    // K offset 0 or 2
  const size_t base = (size_t)b * NOBJ * L + (size_t)row * L;

  v8f accP = {}, accPos = {}, accNeg = {}, accFp = {}, accFn = {};
  float s_prob = 0.f, s_neg = 0.f, s_fneg = 0.f, s_tgt = 0.f;

  // 8 waves stride the L dimension in K=4 steps
  for (int l0 = wave * 4; l0 < L; l0 += 32) {
    const size_t off = base + (size_t)(l0 + khalf);
    v2f x = *(const v2f*)(mask_pred + off);
    v2f t = *(const v2f*)(mask_tgt + off);

    v2f aP, aPos, aNeg, aFp, aFn;
#pragma unroll
    for (int j = 0; j < 2; ++j) {
      float xv   = x[j];
      // neg = softplus(x), pos = softplus(-x) = neg - x  (stable form)
      float negv = fmaxf(xv, 0.0f) + log1pf(__expf(-fabsf(xv)));
      float posv = negv - xv;
      float prob = __expf(-posv);        // sigmoid(x)
      float omp  = __expf(-negv);        // 1 - sigmoid(x)
      float fposv = omp * omp * posv;    // (1-p)^2 * softplus(-x)
      float fnegv = prob * prob * negv;  // p^2 * softplus(x)
      aP[j] = prob; aPos[j] = posv; aNeg[j] = negv; aFp[j] = fposv; aFn[j] = fnegv;
      s_prob += prob; s_neg += negv; s_fneg += fnegv; s_tgt += t[j];
    }

    accP   = __builtin_amdgcn_wmma_f32_16x16x4_f32(false, aP,   false, t, (short)0, accP,   false, false);
    accPos = __builtin_amdgcn_wmma_f32_16x16x4_f32(false, aPos, false, t, (short)0, accPos, false, false);
    accNeg = __builtin_amdgcn_wmma_f32_16x16x4_f32(false, aNeg, false, t, (short)0, accNeg, false, false);
    accFp  = __builtin_amdgcn_wmma_f32_16x16x4_f32(false, aFp,  false, t, (short)0, accFp,  false, false);
    accFn  = __builtin_amdgcn_wmma_f32_16x16x4_f32(false, aFn,  false, t, (short)0, accFn,  false, false);
  }

  // cross-wave reduction: C layout is (M = vgpr + 8*(lane>>4), N = lane&15)
  {
    const int Mbase = (lane >> 4) * 8;
    const int N = lane & 15;
#pragma unroll
    for (int i = 0; i < 8; ++i) {
      int idx = (Mbase + i) * 16 + N;
      atomicAdd(&sMat[0][idx], accP[i]);
      atomicAdd(&sMat[1][idx], accPos[i]);
      atomicAdd(&sMat[2][idx], accNeg[i]);
      atomicAdd(&sMat[3][idx], accFp[i]);
      atomicAdd(&sMat[4][idx], accFn[i]);
    }
    atomicAdd(&sVec[0][row], s_prob);
    atomicAdd(&sVec[1][row], s_neg);
    atomicAdd(&sVec[2][row], s_fneg);
    atomicAdd(&sVec[3][row], s_tgt);
  }
  __syncthreads();

  // finalize: one thread per (n, m)
  {
    const int n = tid >> 4, m = tid & 15;
    float Sp   = sMat[0][tid];
    float Spos = sMat[1][tid];
    float Sneg = sMat[2][tid];
    float Sfp  = sMat[3][tid];
    float Sfn  = sMat[4][tid];
    const float invL = 1.0f / (float)L;

    float dice  = 1.0f - (2.0f * Sp + 1.0f) / (sVec[0][n] + sVec[3][m] + 1.0f);
    float ce    = (Spos + sVec[1][n] - Sneg) * invL;   // pos@tgt + neg@(1-tgt)
    float focal = (Sfp  + sVec[2][n] - Sfn ) * invL;

    int length = 0;
#pragma unroll
    for (int j = 0; j < NOBJ; ++j) length += (sOC[j] != NUM_CLASSES) ? 1 : 0;

    float cls = 0.0f;
    if (m < NUM_CLASSES) {
      int  c  = sOC[m];
      bool vm = (c != NUM_CLASSES);
      int  ci = vm ? c : 0;
      cls = -(class_probs[((size_t)b * NOBJ + n) * (NUM_CLASSES + 1) + ci] * (vm ? 1.0f : 0.0f));
    }

    bool validm = (sOC[m] != NUM_CLASSES);
    float mae = 0.0f;
#pragma unroll
    for (int r = 0; r < 4; ++r) {
      float tp = reg_pred[((size_t)b * NOBJ + n) * 4 + r];
      float tt = reg_tgt [((size_t)b * NOBJ + m) * 4 + r] * (validm ? 1.0f : 0.0f);
      mae += fabsf(tp - tt);
    }
    mae *= 0.25f;

    float cost = cls + dice + ce + focal + mae;
    if (m >= length) cost = __builtin_nanf("");
    out_cost[(size_t)b * (NOBJ * NOBJ) + tid] = cost;
    if (tid == 0) out_len[b] = (float)length;
  }
}

extern "C" void kernel_launch(void* const* d_in, const int* in_sizes, int n_in,
                              void* d_out, int out_size, void* d_ws, size_t ws_size,
                              hipStream_t stream) {
  const float* class_probs  = (const float*)d_in[0];
  const float* mask_pred    = (const float*)d_in[1];
  const float* mask_tgt     = (const float*)d_in[2];
  const float* reg_pred     = (const float*)d_in[3];
  const float* reg_tgt      = (const float*)d_in[4];
  const int*   object_class = (const int*)d_in[5];

  const int B = in_sizes[5] / NOBJ;                 // 2048
  const int L = in_sizes[1] / (B * NOBJ);           // 2048

  float* out_cost = (float*)d_out;                  // B*16*16 floats
  float* out_len  = out_cost + (size_t)B * NOBJ * NOBJ;

  hungarian_cost_kernel<<<B, 256, 0, stream>>>(
      class_probs, mask_pred, mask_tgt, reg_pred, reg_tgt, object_class,
      out_cost, out_len, L);
}